// Graphsynergy_71150428226171
// MI455X (gfx1250) — compile-verified
//
#include <hip/hip_runtime.h>
#include <hip/hip_bf16.h>
#include <cstddef>

// GraphSynergy forward for MI455X (gfx1250, wave32).
// Memory-bound: ~805MB of protein_emb gathers -> ~35us floor @ 23.3 TB/s.
// Dense GEMMs ((3B,128)@(128,64) and (B,128)@(128,64)) use V_WMMA_F32_16X16X4_F32.

typedef __attribute__((ext_vector_type(2))) float v2f;
typedef __attribute__((ext_vector_type(8))) float v8f;

#define BDIM 4096  // not relied upon; B passed at runtime

// ---------------------------------------------------------------------------
// Kernel 0: zero the two scalar accumulators (node_reg, item_reg)
// ---------------------------------------------------------------------------
__global__ void init_acc_kernel(float* acc) {
    if (threadIdx.x < 2) acc[threadIdx.x] = 0.0f;
}

// ---------------------------------------------------------------------------
// Kernel 1: gather neighbors + 2-hop attention aggregation (the 805MB stage).
// grid = (B, 3), block = 128 threads (4 waves).
// Writes Xagg[(p*B + b)*128 + hop*64 + d] and accumulates node sumsq.
// ---------------------------------------------------------------------------
__global__ __launch_bounds__(128) void gather_attn_kernel(
    const int* __restrict__ drug1, const int* __restrict__ drug2,
    const int* __restrict__ cell,
    const int* __restrict__ d1n, const int* __restrict__ d2n,
    const int* __restrict__ cn,
    const float* __restrict__ protein, const float* __restrict__ drug_emb,
    const float* __restrict__ cell_emb,
    float* __restrict__ Xagg, float* __restrict__ node_acc, int B)
{
    __shared__ float ne[128 * 65];   // 128x64 tile, pad to 65 -> conflict-free
    __shared__ float ev[64];         // current item embedding e
    __shared__ float sc[128];        // scores / exp-weights
    __shared__ int   nidx[128];      // neighbor indices for this hop
    __shared__ float red[128];       // sumsq reduction scratch

    const int b = blockIdx.x;
    const int p = blockIdx.y;
    const int t = threadIdx.x;

    const int*   items = (p == 0) ? drug1 : (p == 1) ? drug2 : cell;
    const int*   nbrs  = (p == 0) ? d1n   : (p == 1) ? d2n   : cn;
    const float* table = (p == 2) ? cell_emb : drug_emb;

    const int item = items[b];
    if (t < 64) ev[t] = table[(size_t)item * 64 + t];

    float ssq = 0.0f;
    const size_t nbase = (size_t)b * 2u * 128u;

    for (int hop = 0; hop < 2; ++hop) {
        nidx[t] = nbrs[nbase + (size_t)hop * 128 + t];
        __syncthreads();

        // Gather 128 rows x 256B, 4 threads per row (coalesced float4).
        const int sub = t & 3;
        const int c0  = sub * 16;
        for (int base = 0; base < 128; base += 32) {
            const int m = base + (t >> 2);
            const float4* src =
                reinterpret_cast<const float4*>(protein + (size_t)nidx[m] * 64 + c0);
#pragma unroll
            for (int i = 0; i < 4; ++i) {
                float4 v = src[i];
                float* dst = &ne[m * 65 + c0 + i * 4];
                dst[0] = v.x; dst[1] = v.y; dst[2] = v.z; dst[3] = v.w;
                ssq += v.x * v.x + v.y * v.y + v.z * v.z + v.w * v.w;
            }
        }
        __syncthreads();

        // scores: s[t] = dot(ne[t,:], e)   (reads (t+d)%64 banks -> no conflict)
        float s = 0.0f;
#pragma unroll 8
        for (int d = 0; d < 64; ++d) s += ne[t * 65 + d] * ev[d];
        sc[t] = s;
        __syncthreads();

        // softmax over M=128 (every thread scans LDS broadcast reads)
        float mx = -3.402823466e38f;
        for (int m = 0; m < 128; ++m) mx = fmaxf(mx, sc[m]);
        const float w = __expf(s - mx);
        __syncthreads();
        sc[t] = w;
        __syncthreads();
        float sum = 0.0f;
        for (int m = 0; m < 128; ++m) sum += sc[m];

        // weighted sum: e_new[d] = sum_m w[m] * ne[m][d] / sum
        float enew = 0.0f;
        if (t < 64) {
#pragma unroll 4
            for (int m = 0; m < 128; ++m) enew += sc[m] * ne[m * 65 + t];
            enew /= sum;
            Xagg[((size_t)p * B + b) * 128 + hop * 64 + t] = enew;
        }
        __syncthreads();
        if (t < 64) ev[t] = enew;
        __syncthreads();
    }

    // block-reduce sumsq of gathered neighbor embeddings -> node_reg accumulator
    red[t] = ssq;
    __syncthreads();
    for (int off = 64; off > 0; off >>= 1) {
        if (t < off) red[t] += red[t + off];
        __syncthreads();
    }
    if (t == 0) atomicAdd(node_acc, red[0]);
}

// ---------------------------------------------------------------------------
// Kernel 2: f32 WMMA GEMM  Y(rows x 64) = X(rows x 128) @ W(128 x 64) [+ bias]
// One wave per 16-row tile; 4 column tiles of 16; K loop 128/4 = 32 WMMAs/tile.
// A layout (16x4 f32): lanes 0-15 hold K={k,k+1}, lanes 16-31 hold K={k+2,k+3}.
// B layout (4x16 f32): VGPR0 = rows {k, k+2}, VGPR1 = rows {k+1, k+3}, N = lane%16.
// C/D layout: VGPR r -> M = r + 8*(lane/16), N = lane%16.
// mode 0 (agg):     row = p*B + b; p<2 -> dcat[b*128 + p*64 + n], p==2 -> cbuf[b*64+n]
// mode 1 (combine): combined[row*64 + n]
// ---------------------------------------------------------------------------
__global__ __launch_bounds__(256) void gemm_wmma_kernel(
    const float* __restrict__ X, const float* __restrict__ W,
    const float* __restrict__ bias,
    float* __restrict__ dcat, float* __restrict__ cbuf,
    float* __restrict__ combined, int mode, int B)
{
    const int lane   = threadIdx.x & 31;
    const int waveId = blockIdx.x * (blockDim.x >> 5) + (threadIdx.x >> 5);
    const int row0   = waveId * 16;

    const int half = lane >> 4;   // 0: K pair {0,1}, 1: K pair {2,3}
    const int l16  = lane & 15;

    v8f acc0 = {}, acc1 = {}, acc2 = {}, acc3 = {};

    for (int k = 0; k < 128; k += 4) {
        const int kb = k + half * 2;

        v2f a;
        const float* xr = X + (size_t)(row0 + l16) * 128 + kb;
        a.x = xr[0];
        a.y = xr[1];

        v2f b0, b1, b2, b3;
        const float* w0 = W + (size_t)kb * 64;        // row kb
        const float* w1 = W + (size_t)(kb + 1) * 64;  // row kb+1
        b0.x = w0[l16 +  0]; b0.y = w1[l16 +  0];
        b1.x = w0[l16 + 16]; b1.y = w1[l16 + 16];
        b2.x = w0[l16 + 32]; b2.y = w1[l16 + 32];
        b3.x = w0[l16 + 48]; b3.y = w1[l16 + 48];

        acc0 = __builtin_amdgcn_wmma_f32_16x16x4_f32(false, a, false, b0,
                                                     (short)0, acc0, false, false);
        acc1 = __builtin_amdgcn_wmma_f32_16x16x4_f32(false, a, false, b1,
                                                     (short)0, acc1, false, false);
        acc2 = __builtin_amdgcn_wmma_f32_16x16x4_f32(false, a, false, b2,
                                                     (short)0, acc2, false, false);
        acc3 = __builtin_amdgcn_wmma_f32_16x16x4_f32(false, a, false, b3,
                                                     (short)0, acc3, false, false);
    }

    v8f accs[4] = {acc0, acc1, acc2, acc3};
#pragma unroll
    for (int r = 0; r < 8; ++r) {
        const int m    = r + half * 8;
        const int grow = row0 + m;
#pragma unroll
        for (int tile = 0; tile < 4; ++tile) {
            const int n = tile * 16 + l16;
            float v = accs[tile][r];
            if (bias) v += bias[n];
            if (mode == 0) {
                const int p = grow / B;
                const int b = grow - p * B;
                if (p < 2) dcat[(size_t)b * 128 + p * 64 + n] = v;
                else       cbuf[(size_t)b * 64 + n] = v;
            } else {
                combined[(size_t)grow * 64 + n] = v;
            }
        }
    }
}

// ---------------------------------------------------------------------------
// Kernel 3: per-item therapy/toxic dots, sigmoid, item_reg sumsq.
// One wave32 per batch item; 8 items per 256-thread block.
// ---------------------------------------------------------------------------
__global__ __launch_bounds__(256) void finalize_kernel(
    const float* __restrict__ dcat, const float* __restrict__ cbuf,
    const float* __restrict__ combined,
    float* __restrict__ x_out, float* __restrict__ item_acc, int B)
{
    const int lane = threadIdx.x & 31;
    const int wave = threadIdx.x >> 5;
    const int b    = blockIdx.x * 8 + wave;

    float th = 0.0f, tox = 0.0f, isq = 0.0f;
#pragma unroll
    for (int n = lane; n < 64; n += 32) {
        const float c  = cbuf[(size_t)b * 64 + n];
        const float cm = combined[(size_t)b * 64 + n];
        const float a1 = dcat[(size_t)b * 128 + n];
        const float a2 = dcat[(size_t)b * 128 + 64 + n];
        th  += cm * c;
        tox += a1 * a2;
        isq += c * c + a1 * a1 + a2 * a2;
    }
#pragma unroll
    for (int off = 16; off > 0; off >>= 1) {
        th  += __shfl_down(th,  off, 32);
        tox += __shfl_down(tox, off, 32);
        isq += __shfl_down(isq, off, 32);
    }
    if (lane == 0) {
        x_out[b] = 1.0f / (1.0f + __expf(-(th - tox)));
        atomicAdd(item_acc, isq);
    }
}

// ---------------------------------------------------------------------------
// Kernel 4: emb_loss = 1e-6 * (item_reg/2 + node_reg/2) / B
// ---------------------------------------------------------------------------
__global__ void loss_kernel(const float* __restrict__ acc, float* __restrict__ out,
                            int B) {
    if (threadIdx.x == 0 && blockIdx.x == 0) {
        const float node_reg = acc[0] * 0.5f;
        const float item_reg = acc[1] * 0.5f;
        out[0] = 1e-6f * (item_reg + node_reg) / (float)B;
    }
}

// ---------------------------------------------------------------------------
extern "C" void kernel_launch(void* const* d_in, const int* in_sizes, int n_in,
                              void* d_out, int out_size, void* d_ws, size_t ws_size,
                              hipStream_t stream) {
    const int B = in_sizes[0];  // 4096

    const int*   drug1     = (const int*)d_in[0];
    const int*   drug2     = (const int*)d_in[1];
    const int*   cell      = (const int*)d_in[2];
    const int*   d1n       = (const int*)d_in[3];
    const int*   d2n       = (const int*)d_in[4];
    const int*   cn        = (const int*)d_in[5];
    const float* protein   = (const float*)d_in[6];
    const float* drug_emb  = (const float*)d_in[7];
    const float* cell_emb  = (const float*)d_in[8];
    const float* combine_W = (const float*)d_in[9];
    const float* agg_W     = (const float*)d_in[10];
    const float* agg_b     = (const float*)d_in[11];

    // workspace layout (floats)
    float* ws       = (float*)d_ws;
    float* Xagg     = ws;                                 // 3*B*128
    float* dcat     = Xagg + (size_t)3 * B * 128;         // B*128  ([d1|d2] per row)
    float* cbuf     = dcat + (size_t)B * 128;             // B*64
    float* combined = cbuf + (size_t)B * 64;              // B*64
    float* acc      = combined + (size_t)B * 64;          // [0]=node_reg, [1]=item_reg

    float* x_out    = (float*)d_out;                      // B floats
    float* loss_out = x_out + B;                          // 1 float

    init_acc_kernel<<<1, 32, 0, stream>>>(acc);

    gather_attn_kernel<<<dim3(B, 3), 128, 0, stream>>>(
        drug1, drug2, cell, d1n, d2n, cn, protein, drug_emb, cell_emb,
        Xagg, acc /*node*/, B);

    // agg GEMM: (3B x 128) @ agg_W + agg_b -> dcat / cbuf
    const int aggTiles = (3 * B) / 16;            // 768 waves
    gemm_wmma_kernel<<<aggTiles / 8, 256, 0, stream>>>(
        Xagg, agg_W, agg_b, dcat, cbuf, nullptr, /*mode=*/0, B);

    // combine GEMM: (B x 128) @ combine_W -> combined
    const int cmbTiles = B / 16;                  // 256 waves
    gemm_wmma_kernel<<<cmbTiles / 8, 256, 0, stream>>>(
        dcat, combine_W, nullptr, nullptr, nullptr, combined, /*mode=*/1, B);

    finalize_kernel<<<B / 8, 256, 0, stream>>>(
        dcat, cbuf, combined, x_out, acc + 1 /*item*/, B);

    loss_kernel<<<1, 32, 0, stream>>>(acc, loss_out, B);
}